// MSDeformableAttention_55164559950056
// MI455X (gfx1250) — compile-verified
//
#include <hip/hip_runtime.h>
#include <hip/hip_bf16.h>

typedef __attribute__((ext_vector_type(2))) float v2f;
typedef __attribute__((ext_vector_type(4))) float v4f;
typedef __attribute__((ext_vector_type(8))) float v8f;

#define MODE_PLAIN 0
#define MODE_V     1
#define MODE_LOC   2

static __device__ __forceinline__ v8f wmma_k4(v2f a, v2f b, v8f c) {
  // D = A(16x4 f32) * B(4x16 f32) + C(16x16 f32)
  return __builtin_amdgcn_wmma_f32_16x16x4_f32(false, a, false, b, (short)0, c, false, false);
}

// Generic fp32 WMMA GEMM: C[M,N] = A[M,K] @ B[K,N] (+bias), epilogue per `mode`.
// Block = 128 threads (4 waves), 64x64 C tile; wave w owns rows w*16..w*16+15,
// all 64 cols (four 16x16 WMMA accumulators). K in chunks of 32 through LDS.
__launch_bounds__(128)
__global__ void gemm_wmma_f32(const float* __restrict__ A, const float* __restrict__ B,
                              const float* __restrict__ bias, float* __restrict__ out,
                              int M, int N, int K, int mode,
                              const float* __restrict__ ref, float* __restrict__ loc_out)
{
  __shared__ float As[64][34];   // pad to kill bank conflicts on column-pair reads
  __shared__ float Bs[32][66];
  const int tid   = threadIdx.x;
  const int wave  = tid >> 5;
  const int lane  = tid & 31;
  const int l16   = lane & 15;
  const int lhalf = lane >> 4;
  const int row0  = blockIdx.y * 64;
  const int col0  = blockIdx.x * 64;
  const int wrow  = row0 + (wave << 4);

  v8f acc[4];
#pragma unroll
  for (int t = 0; t < 4; ++t)
    acc[t] = (v8f){0.f, 0.f, 0.f, 0.f, 0.f, 0.f, 0.f, 0.f};

  for (int kc = 0; kc < K; kc += 32) {
    // Stage A: 64x32 floats (512 float4, 128 threads x 4)
#pragma unroll
    for (int i = 0; i < 4; ++i) {
      int e = tid + i * 128;
      int r = e >> 3;
      int c = (e & 7) << 2;
      v4f val = *(const v4f*)(A + (size_t)(row0 + r) * K + kc + c);
      As[r][c + 0] = val.x; As[r][c + 1] = val.y;
      As[r][c + 2] = val.z; As[r][c + 3] = val.w;
    }
    // Stage B: 32x64 floats
#pragma unroll
    for (int i = 0; i < 4; ++i) {
      int e = tid + i * 128;
      int r = e >> 4;
      int c = (e & 15) << 2;
      v4f val = *(const v4f*)(B + (size_t)(kc + r) * N + col0 + c);
      Bs[r][c + 0] = val.x; Bs[r][c + 1] = val.y;
      Bs[r][c + 2] = val.z; Bs[r][c + 3] = val.w;
    }
    __syncthreads();
#pragma unroll
    for (int kk = 0; kk < 32; kk += 4) {
      int ka = kk + (lhalf << 1);   // lanes 0-15: K=kk,kk+1 ; lanes 16-31: K=kk+2,kk+3
      v2f af = *(const v2f*)&As[(wave << 4) + l16][ka];
#pragma unroll
      for (int t = 0; t < 4; ++t) {
        v2f bf;
        bf.x = Bs[ka][(t << 4) + l16];
        bf.y = Bs[ka + 1][(t << 4) + l16];
        acc[t] = wmma_k4(af, bf, acc[t]);
      }
    }
    __syncthreads();
  }

  // Epilogue. C layout: VGPR i, lanes 0-15 -> M=i, N=lane; lanes 16-31 -> M=i+8.
  if (mode == MODE_PLAIN) {
#pragma unroll
    for (int t = 0; t < 4; ++t) {
      int col = col0 + (t << 4) + l16;
      float bv = bias[col];
#pragma unroll
      for (int i = 0; i < 8; ++i) {
        int row = wrow + i + (lhalf << 3);
        out[(size_t)row * N + col] = acc[t][i] + bv;
      }
    }
  } else if (mode == MODE_V) {
    // scatter into v[b][h][lv][32]; row = b*5440+lv, col = h*32+d
#pragma unroll
    for (int t = 0; t < 4; ++t) {
      int col = col0 + (t << 4) + l16;
      float bv = bias[col];
      int h = col >> 5;
      int d = col & 31;
#pragma unroll
      for (int i = 0; i < 8; ++i) {
        int row = wrow + i + (lhalf << 3);
        int b  = row / 5440;
        int lv = row - b * 5440;
        out[(((size_t)(b * 8 + h)) * 5440 + lv) * 32 + d] = acc[t][i] + bv;
      }
    }
  } else { // MODE_LOC: fused sampling-location epilogue
    // col n = (((h*4+l)*16+p)*2+xy); loc = poly(ref,p,xy) + (acc+bias)/size[l]
#pragma unroll
    for (int t = 0; t < 4; ++t) {
      int n = col0 + (t << 4) + l16;
      float bv = bias[n];
      int lvl = (n >> 5) & 3;
      int p   = (n >> 1) & 15;
      int xy  = n & 1;
      float inv_norm = 1.0f / (float)(64 >> lvl);
      float lam = (float)p * (1.0f / 15.0f);
#pragma unroll
      for (int i = 0; i < 8; ++i) {
        int row = wrow + i + (lhalf << 3);
        v4f rp = *(const v4f*)(ref + (size_t)row * 8 + (xy << 2));
        float pnt = ((rp.x * lam + rp.y) * lam + rp.z) * lam + rp.w;
        loc_out[(size_t)row * 1024 + n] = pnt + (acc[t][i] + bv) * inv_norm;
      }
    }
  }
}

// One wave per (query-row, head): softmax over 64 logits (2 per lane).
__launch_bounds__(256)
__global__ void softmax64_kernel(const float* __restrict__ logits, float* __restrict__ attn)
{
  const int wave = threadIdx.x >> 5;
  const int lane = threadIdx.x & 31;
  const int gid  = blockIdx.x * 8 + wave;            // gid = row*8 + h, row < 4096
  const float* src = logits + ((size_t)(gid >> 3)) * 512 + (size_t)(gid & 7) * 64;
  float a0 = src[lane];
  float a1 = src[lane + 32];
  float m = fmaxf(a0, a1);
#pragma unroll
  for (int off = 16; off > 0; off >>= 1) m = fmaxf(m, __shfl_xor(m, off, 32));
  float e0 = __expf(a0 - m);
  float e1 = __expf(a1 - m);
  float s = e0 + e1;
#pragma unroll
  for (int off = 16; off > 0; off >>= 1) s += __shfl_xor(s, off, 32);
  float inv = 1.0f / s;
  float* dst = attn + (size_t)gid * 64;
  dst[lane]      = e0 * inv;
  dst[lane + 32] = e1 * inv;
}

// One wave per (b,q,h); lane = channel d (0..31). Bilinear gather from
// head-major v[b][h][lv][32] (L2-resident), weighted by attn.
__launch_bounds__(256)
__global__ void deform_gather_kernel(const float* __restrict__ v, const float* __restrict__ loc,
                                     const float* __restrict__ attn, float* __restrict__ pre)
{
  const int wave = threadIdx.x >> 5;
  const int lane = threadIdx.x & 31;
  const int gid  = blockIdx.x * 8 + wave;   // gid = m*8 + h
  const int m = gid >> 3;
  const int b = m >> 10;
  const int h = gid & 7;
  const float* vb   = v + ((size_t)(b * 8 + h)) * 5440 * 32;
  const float* locp = loc + (size_t)gid * 128;
  const float* ap   = attn + (size_t)gid * 64;
  float sum = 0.0f;
  int start = 0;
#pragma unroll
  for (int l = 0; l < 4; ++l) {
    const int S = 64 >> l;
    const float fS = (float)S;
    const float* vl = vb + (size_t)start * 32;
    for (int p = 0; p < 16; ++p) {
      float lx = locp[(((l << 4) + p) << 1) + 0];
      float ly = locp[(((l << 4) + p) << 1) + 1];
      float aw = ap[(l << 4) + p];
      float x = lx * fS - 0.5f;
      float y = ly * fS - 0.5f;
      float x0f = floorf(x);
      float y0f = floorf(y);
      float wx = x - x0f;
      float wy = y - y0f;
      int x0 = (int)x0f;
      int y0 = (int)y0f;
#pragma unroll
      for (int c = 0; c < 4; ++c) {
        int dx = c & 1;
        int dy = c >> 1;
        int xi = x0 + dx;
        int yi = y0 + dy;
        float w = (dx ? wx : 1.0f - wx) * (dy ? wy : 1.0f - wy);
        bool valid = (xi >= 0) & (xi < S) & (yi >= 0) & (yi < S);
        int xc = xi < 0 ? 0 : (xi >= S ? S - 1 : xi);
        int yc = yi < 0 ? 0 : (yi >= S ? S - 1 : yi);
        float g = vl[((size_t)(yc * S + xc)) * 32 + lane];
        sum += aw * (valid ? w : 0.0f) * g;
      }
    }
    start += S * S;
  }
  pre[(size_t)gid * 32 + lane] = sum;
}

extern "C" void kernel_launch(void* const* d_in, const int* in_sizes, int n_in,
                              void* d_out, int out_size, void* d_ws, size_t ws_size,
                              hipStream_t stream) {
  (void)in_sizes; (void)n_in; (void)out_size; (void)ws_size;
  const float* query   = (const float*)d_in[0];
  const float* refpts  = (const float*)d_in[1];
  const float* value   = (const float*)d_in[2];
  const float* W_value = (const float*)d_in[3];
  const float* b_value = (const float*)d_in[4];
  const float* W_off   = (const float*)d_in[5];
  const float* b_off   = (const float*)d_in[6];
  const float* W_attn  = (const float*)d_in[7];
  const float* b_attn  = (const float*)d_in[8];
  const float* W_out   = (const float*)d_in[9];
  const float* b_out   = (const float*)d_in[10];

  float* out       = (float*)d_out;
  float* out_final = out;                        // [4,1024,256]       = 1,048,576
  float* out_loc   = out + 1048576;              // [4,1024,8,4,16,2]  = 4,194,304
  float* out_attn  = out + 1048576 + 4194304;    // [4,1024,8,4,16]    = 2,097,152

  float* ws     = (float*)d_ws;
  float* v      = ws;                 // [4,8,5440,32] = 5,570,560 floats
  float* logits = ws + 5570560;       // [4096,512]    = 2,097,152 floats
  float* pre    = ws + 7667712;       // [4096,256]    = 1,048,576 floats

  // 1) v = value @ W_value + b_value, scattered to [b,h,lv,hd]
  gemm_wmma_f32<<<dim3(256 / 64, 21760 / 64), 128, 0, stream>>>(
      value, W_value, b_value, v, 21760, 256, 256, MODE_V, nullptr, nullptr);
  // 2) offsets GEMM fused into sampling_locations (writes d_out directly)
  gemm_wmma_f32<<<dim3(1024 / 64, 4096 / 64), 128, 0, stream>>>(
      query, W_off, b_off, nullptr, 4096, 1024, 256, MODE_LOC, refpts, out_loc);
  // 3) attention logits
  gemm_wmma_f32<<<dim3(512 / 64, 4096 / 64), 128, 0, stream>>>(
      query, W_attn, b_attn, logits, 4096, 512, 256, MODE_PLAIN, nullptr, nullptr);
  // 4) softmax over 64 per (q,h) -> d_out attn segment
  softmax64_kernel<<<4096, 256, 0, stream>>>(logits, out_attn);
  // 5) bilinear gather + attn-weighted accumulate
  deform_gather_kernel<<<4096, 256, 0, stream>>>(v, out_loc, out_attn, pre);
  // 6) output projection
  gemm_wmma_f32<<<dim3(256 / 64, 4096 / 64), 128, 0, stream>>>(
      pre, W_out, b_out, out_final, 4096, 256, 256, MODE_PLAIN, nullptr, nullptr);
}